// SVGD_23665269801093
// MI455X (gfx1250) — compile-verified
//
#include <hip/hip_runtime.h>

typedef __attribute__((ext_vector_type(2))) float v2f;
typedef __attribute__((ext_vector_type(8))) float v8f;

#define NP 2048
#define ND 32
#define NB 4096                 // histogram bins per refinement level
#define NN (NP * NP)
#define NSPLIT 8                // j-dimension split for the SVGD GEMM
#define JCHUNK (NP / NSPLIT)    // 256 columns per chunk

// workspace layout (4-byte units)                                   bytes
#define OFF_D2   0                              // NP*NP floats      16 MB (L2-resident)
#define OFF_S    (NP * NP)                      // NP floats
#define OFF_ST   (OFF_S + NP)                   // 16 words state
#define OFF_HIST (OFF_ST + 16)                  // NB uints
#define OFF_KXP  (OFF_HIST + NB)                // NSPLIT*NP*ND floats  2 MB
#define OFF_RSP  (OFF_KXP + NSPLIT * NP * ND)   // NSPLIT*NP floats    64 KB

// state word indices
#define ST_MAXBITS 0
#define ST_LO0 1
#define ST_HI0 2
#define ST_BASE0 3
#define ST_RES0 4
#define ST_LO1 5
#define ST_HI1 6
#define ST_BASE1 7
#define ST_RES1 8
#define ST_INVH 9

static __device__ __forceinline__ v8f wmma_f32(v2f a, v2f b, v8f c) {
    // V_WMMA_F32_16X16X4_F32 : D(16x16,f32) = A(16x4,f32) * B(4x16,f32) + C
    return __builtin_amdgcn_wmma_f32_16x16x4_f32(false, a, false, b, (short)0, c,
                                                 false, false);
}

// ---- zero histogram + state ------------------------------------------------
__global__ void k_zero(unsigned* __restrict__ hist, unsigned* __restrict__ stu) {
    int t = blockIdx.x * blockDim.x + threadIdx.x;
    for (int i = t; i < NB; i += gridDim.x * blockDim.x) hist[i] = 0u;
    if (t < 16) stu[t] = 0u;
}

// ---- s[i] = ||x_i||^2 (one wave32 per row, D==32) --------------------------
__global__ void k_sq(const float* __restrict__ x, float* __restrict__ s) {
    int wid  = (blockIdx.x * blockDim.x + threadIdx.x) >> 5;
    int lane = threadIdx.x & 31;
    if (wid >= NP) return;
    float v = x[wid * ND + lane];
    float p = v * v;
    #pragma unroll
    for (int off = 16; off > 0; off >>= 1) p += __shfl_xor(p, off);
    if (lane == 0) s[wid] = p;
}

// ---- D2[i,j] = s_i + s_j - 2*x_i.x_j via f32 WMMA Gram tiles ---------------
__global__ void k_gram(const float* __restrict__ x, const float* __restrict__ s,
                       float* __restrict__ d2) {
    int wid  = (blockIdx.x * blockDim.x + threadIdx.x) >> 5;  // one 16x16 tile per wave
    int lane = threadIdx.x & 31;
    int i0 = (wid >> 7) << 4;        // tile row
    int j0 = (wid & 127) << 4;       // tile col
    int h = lane >> 4, n = lane & 15;

    const float2* xa = (const float2*)(x + (size_t)(i0 + n) * ND);
    const float2* xb = (const float2*)(x + (size_t)(j0 + n) * ND);
    v8f g = {};
    #pragma unroll
    for (int kk = 0; kk < 8; ++kk) {      // K = 32 = 8 steps of 4
        float2 fa = xa[2 * kk + h];       // A[M=n][K=4kk+2h, +1]
        float2 fb = xb[2 * kk + h];       // B[K=4kk+2h, +1][N=n]
        v2f a; a.x = fa.x; a.y = fa.y;
        v2f b; b.x = fb.x; b.y = fb.y;
        g = wmma_f32(a, b, g);
    }
    float sj = s[j0 + n];
    #pragma unroll
    for (int v = 0; v < 8; ++v) {         // D layout: M = v + 8h, N = n
        int m = v + 8 * h;
        float val = s[i0 + m] + sj - 2.0f * g[v];
        d2[(size_t)(i0 + m) * NP + (j0 + n)] = fmaxf(val, 0.0f);
    }
}

// ---- global max of D2 (for histogram range) --------------------------------
__global__ void k_max(const float* __restrict__ d2, unsigned* __restrict__ stu) {
    size_t t = blockIdx.x * blockDim.x + threadIdx.x;
    float m = 0.0f;
    for (size_t i = t; i < (size_t)NN; i += (size_t)gridDim.x * blockDim.x)
        m = fmaxf(m, d2[i]);
    #pragma unroll
    for (int off = 16; off > 0; off >>= 1) m = fmaxf(m, __shfl_xor(m, off));
    if ((threadIdx.x & 31) == 0)
        atomicMax(&stu[ST_MAXBITS], __float_as_uint(m));   // valid: all values >= 0
}

__global__ void k_initsel(unsigned* __restrict__ stu) {
    float mx = __uint_as_float(stu[ST_MAXBITS]) * 1.0000005f + 1e-20f;
    float* stf = (float*)stu;
    stf[ST_LO0] = 0.0f; stf[ST_HI0] = mx; stu[ST_BASE0] = 0u;
    stf[ST_LO1] = 0.0f; stf[ST_HI1] = mx; stu[ST_BASE1] = 0u;
}

// ---- one histogram pass over D2 in [lo,hi) ---------------------------------
__global__ void k_hist(const float* __restrict__ d2, const unsigned* __restrict__ stu,
                       unsigned* __restrict__ hist, int chain) {
    __shared__ unsigned lh[NB];
    for (int i = threadIdx.x; i < NB; i += blockDim.x) lh[i] = 0u;
    __syncthreads();
    const float* stf = (const float*)stu;
    float lo = stf[chain ? ST_LO1 : ST_LO0];
    float hi = stf[chain ? ST_HI1 : ST_HI0];
    float scale = (float)NB / (hi - lo);
    size_t stride = (size_t)gridDim.x * blockDim.x;
    for (size_t i = blockIdx.x * blockDim.x + threadIdx.x; i < (size_t)NN; i += stride) {
        float v = d2[i];
        if (v >= lo && v < hi) {
            int b = (int)((v - lo) * scale);
            if (b > NB - 1) b = NB - 1;
            atomicAdd(&lh[b], 1u);
        }
    }
    __syncthreads();
    for (int i = threadIdx.x; i < NB; i += blockDim.x)
        if (lh[i]) atomicAdd(&hist[i], lh[i]);
}

// ---- select bin containing target rank, narrow [lo,hi) ---------------------
__global__ void k_scan(unsigned* __restrict__ stu, unsigned* __restrict__ hist,
                       int chain, unsigned rank, int zero_after) {
    float* stf = (float*)stu;
    int iLo = chain ? ST_LO1 : ST_LO0, iHi = chain ? ST_HI1 : ST_HI0;
    int iB  = chain ? ST_BASE1 : ST_BASE0, iR = chain ? ST_RES1 : ST_RES0;
    float lo = stf[iLo], hi = stf[iHi];
    float w = (hi - lo) / (float)NB;
    unsigned cum = stu[iB];
    int sel = NB - 1;
    for (int b = 0; b < NB; ++b) {
        unsigned nb = cum + hist[b];
        if (rank < nb) { sel = b; break; }
        cum = nb;
    }
    stf[iLo] = lo + sel * w;
    stf[iHi] = lo + (sel + 1) * w;
    stu[iB]  = cum;
    stf[iR]  = lo + sel * w;
    if (zero_after)
        for (int b = 0; b < NB; ++b) hist[b] = 0u;
}

// ---- h = median(dist)/ln(N); store 1/(h+1e-6) ------------------------------
__global__ void k_finish(unsigned* __restrict__ stu) {
    float* stf = (float*)stu;
    float med = 0.5f * (sqrtf(stf[ST_RES0]) + sqrtf(stf[ST_RES1]));  // avg of two middles
    float h = med / logf((float)NP);
    stf[ST_INVH] = 1.0f / (h + 1e-6f);
}

// ---- partial fused SVGD GEMM over one j-chunk ------------------------------
// grid: NSPLIT j-chunks x 16 blocks; block = 256 thr = 8 waves = 8 i-blocks.
// Each wave owns a (jc, i0) pair -> deterministic non-atomic partial stores.
__global__ void __launch_bounds__(256)
k_svgd_part(const float* __restrict__ x, const float* __restrict__ d2,
            const unsigned* __restrict__ stu,
            float* __restrict__ kxp, float* __restrict__ rsp) {
    __shared__ float xs[JCHUNK * ND];                         // 32 KB of 320 KB WGP LDS

    int jc   = blockIdx.x & (NSPLIT - 1);                     // which j-chunk
    int ibb  = (blockIdx.x >> 3) << 3;                        // first i-block of this block
    int wave = threadIdx.x >> 5;
    int lane = threadIdx.x & 31;
    int i0   = (ibb + wave) << 4;
    int h = lane >> 4, n = lane & 15;

    // cooperative stage of x[jc*JCHUNK .. +JCHUNK) into LDS (B operand)
    const float4* xg = (const float4*)(x + (size_t)jc * JCHUNK * ND);
    float4* xs4 = (float4*)xs;
    for (int i = threadIdx.x; i < JCHUNK * ND / 4; i += blockDim.x) xs4[i] = xg[i];
    __syncthreads();

    float invh = ((const float*)stu)[ST_INVH];
    v8f acc0 = {}, acc1 = {};          // partial K@x, feature cols 0..15 / 16..31
    float rs = 0.0f;                   // partial rowsum(K) for row i0+n
    const float2* drow =
        (const float2*)(d2 + (size_t)(i0 + n) * NP + (size_t)jc * JCHUNK);

    for (int j = 0; j < JCHUNK; j += 4) {
        float2 dv = drow[(j >> 1) + h];                   // D2[row][jc*JCHUNK+j+2h, +1]
        v2f a; a.x = __expf(-dv.x * invh);                // A fragment = K tile 16x4
               a.y = __expf(-dv.y * invh);
        rs += a.x + a.y;
        int jr = j + 2 * h;
        v2f b0, b1;                                       // B[K][N] = xs[j+K][fbase+N]
        b0.x = xs[jr * ND + n];       b0.y = xs[(jr + 1) * ND + n];
        b1.x = xs[jr * ND + 16 + n];  b1.y = xs[(jr + 1) * ND + 16 + n];
        acc0 = wmma_f32(a, b0, acc0);
        acc1 = wmma_f32(a, b1, acc1);
    }
    rs += __shfl_xor(rs, 16);          // full chunk rowsum for row i0+n
    if (h == 0) rsp[jc * NP + i0 + n] = rs;

    #pragma unroll
    for (int v = 0; v < 8; ++v) {      // D layout: M = v + 8h, N = n
        int m = v + 8 * h;
        size_t base = ((size_t)jc * NP + (i0 + m)) * ND;
        kxp[base + n]      = acc0[v];
        kxp[base + 16 + n] = acc1[v];
    }
}

// ---- deterministic reduction over NSPLIT partials + update -----------------
__global__ void k_update(const float* __restrict__ x, const float* __restrict__ kxp,
                         const float* __restrict__ rsp, float* __restrict__ out) {
    int t = blockIdx.x * blockDim.x + threadIdx.x;
    if (t >= NP * ND) return;
    int i = t >> 5;                    // ND == 32
    float kx = 0.0f, rs = 0.0f;
    #pragma unroll
    for (int p = 0; p < NSPLIT; ++p) {
        kx += kxp[(size_t)p * NP * ND + t];
        rs += rsp[p * NP + i];
    }
    float xv = x[t];
    float g = (0.1f / (float)NP) * (rs * xv - 2.0f * kx);
    g = fminf(fmaxf(g, -1000.0f), 1000.0f);
    out[t] = xv + g;
}

extern "C" void kernel_launch(void* const* d_in, const int* in_sizes, int n_in,
                              void* d_out, int out_size, void* d_ws, size_t ws_size,
                              hipStream_t stream) {
    const float* x = (const float*)d_in[0];
    float* out = (float*)d_out;
    float* ws = (float*)d_ws;

    float*    d2   = ws + OFF_D2;
    float*    s    = ws + OFF_S;
    unsigned* stu  = (unsigned*)(ws + OFF_ST);
    unsigned* hist = (unsigned*)(ws + OFF_HIST);
    float*    kxp  = ws + OFF_KXP;
    float*    rsp  = ws + OFF_RSP;

    k_zero<<<16, 256, 0, stream>>>(hist, stu);
    k_sq<<<NP / 8, 256, 0, stream>>>(x, s);                       // 8 waves/block
    k_gram<<<(NP / 16) * (NP / 16) / 8, 256, 0, stream>>>(x, s, d2);
    k_max<<<1024, 256, 0, stream>>>(d2, stu);
    k_initsel<<<1, 1, 0, stream>>>(stu);

    const unsigned r0 = NN / 2 - 1, r1 = NN / 2;   // two middle ranks (0-indexed)
    // level 1: both chains share the same [0,max) histogram
    k_hist<<<512, 256, 0, stream>>>(d2, stu, hist, 0);
    k_scan<<<1, 1, 0, stream>>>(stu, hist, 0, r0, 0);
    k_scan<<<1, 1, 0, stream>>>(stu, hist, 1, r1, 1);
    // levels 2..3 per chain
    for (int it = 0; it < 2; ++it) {
        k_hist<<<512, 256, 0, stream>>>(d2, stu, hist, 0);
        k_scan<<<1, 1, 0, stream>>>(stu, hist, 0, r0, 1);
    }
    for (int it = 0; it < 2; ++it) {
        k_hist<<<512, 256, 0, stream>>>(d2, stu, hist, 1);
        k_scan<<<1, 1, 0, stream>>>(stu, hist, 1, r1, 1);
    }
    k_finish<<<1, 1, 0, stream>>>(stu);

    k_svgd_part<<<NSPLIT * (NP / 16 / 8), 256, 0, stream>>>(x, d2, stu, kxp, rsp);
    k_update<<<NP * ND / 256, 256, 0, stream>>>(x, kxp, rsp, out);
}